// QNetwork_43404939493632
// MI455X (gfx1250) — compile-verified
//
#include <hip/hip_runtime.h>

typedef float v2f __attribute__((ext_vector_type(2)));
typedef float v8f __attribute__((ext_vector_type(8)));

#define D 256
#define MB 4   // m-tiles per wave
#define NB 2   // n-tiles per wave (16/NB = 8 n-blocks = 8 waves = 1 block covers all N)

// ---------------- zero fill ----------------
__global__ void zero_kernel(float* __restrict__ p, int n) {
    int i = blockIdx.x * blockDim.x + threadIdx.x;
    if (i < n) p[i] = 0.0f;
}

// ---- M2 = W2 @ lin_W[:D], M3 = W3 @ lin_W[D:], stored PACKED for WMMA ----
// Packed layout: element (k, n) lives at (k>>1)*2*D + 2*n + (k&1), so the
// per-lane WMMA B operand pair (B[k][n], B[k+1][n]) is one b64 load.
__global__ void premul_pack_kernel(const float* __restrict__ W,
                                   const float* __restrict__ linW,
                                   float* __restrict__ Mout) {
    int i = blockIdx.x;   // k row of result
    int j = threadIdx.x;  // n col of result
    float s = 0.0f;
    #pragma unroll 8
    for (int k = 0; k < D; ++k)
        s += W[i * D + k] * linW[k * D + j];
    Mout[(i >> 1) * (2 * D) + 2 * j + (i & 1)] = s;
}

// ---------------- encode: h = l2norm(relu(X @ W1)), F small ----------------
__global__ void encode_kernel(const float* __restrict__ X,   // [rows, F]
                              const float* __restrict__ W1,  // [F, D]
                              float* __restrict__ Outp,      // [rows, D]
                              int F) {
    int row = blockIdx.x;
    int t   = threadIdx.x;  // 256 threads, one per output column
    float v = 0.0f;
    for (int f = 0; f < F; ++f)
        v += X[row * F + f] * W1[f * D + t];
    v = v > 0.0f ? v : 0.0f;

    __shared__ float red[D];
    red[t] = v * v;
    __syncthreads();
    for (int s = D / 2; s > 0; s >>= 1) {
        if (t < s) red[t] += red[t + s];
        __syncthreads();
    }
    float scale = 1.0f / fmaxf(sqrtf(red[0]), 1e-12f);
    Outp[row * D + t] = v * scale;
}

// ---------------- edge aggregation: h_nv[dst] += w * h[src] ----------------
// One block (256 threads) per edge; gather + atomics stay in L2 (h is 20.5 MB).
__global__ void edge_agg_kernel(const int* __restrict__ src,
                                const int* __restrict__ dst,
                                const float* __restrict__ w,
                                const float* __restrict__ h,
                                float* __restrict__ h_nv) {
    int e = blockIdx.x;
    int t = threadIdx.x;
    int s = src[e];
    int d = dst[e];
    float ww = w[e];
    atomicAdd(&h_nv[d * D + t], ww * h[s * D + t]);
}

// ---------------- batch aggregation: h_nv_s[batch[i]] += h[i] ----------------
__global__ void batch_agg_kernel(const int* __restrict__ ba,
                                 const float* __restrict__ h,
                                 float* __restrict__ h_nv_s) {
    int i = blockIdx.x;
    int t = threadIdx.x;
    atomicAdd(&h_nv_s[ba[i] * D + t], h[i * D + t]);
}

// ------ fused dual GEMM + bias + ReLU via WMMA f32 16x16x4, blocked -------
// Out = relu(A1 @ B1 + A2 @ B2 + bias). B1p/B2p are in packed layout.
// Wave computes MBx16 x NBx16 output; block (8 waves) covers one 64-row
// m-block across all 256 columns.
__global__ __launch_bounds__(256)
void gemm2_relu_kernel(const float* __restrict__ A1,
                       const float* __restrict__ A2,
                       const float* __restrict__ B1p,
                       const float* __restrict__ B2p,
                       const float* __restrict__ bias,
                       float* __restrict__ Outp,
                       int mtiles) {
    const int lane   = threadIdx.x & 31;
    const int wave   = threadIdx.x >> 5;
    const int waveId = blockIdx.x * 8 + wave;
    const int nblk   = waveId & 7;           // 8 n-blocks of NB tiles
    const int mblk   = waveId >> 3;

    const int l15  = lane & 15;
    const int koff = (lane >> 4) << 1;       // 0 for lanes 0-15, 2 for 16-31

    int ncol[NB];
    #pragma unroll
    for (int j = 0; j < NB; ++j)
        ncol[j] = (nblk * NB + j) * 16 + l15;

    int mt[MB];
    const float* a1[MB];
    const float* a2[MB];
    #pragma unroll
    for (int i = 0; i < MB; ++i) {
        int t = mblk * MB + i;
        if (t > mtiles - 1) t = mtiles - 1;  // wave-uniform clamp for loads
        mt[i] = t;
        a1[i] = A1 + (size_t)(t * 16 + l15) * D;
        a2[i] = A2 + (size_t)(t * 16 + l15) * D;
    }

    v8f acc[MB][NB];
    #pragma unroll
    for (int i = 0; i < MB; ++i)
        #pragma unroll
        for (int j = 0; j < NB; ++j)
            acc[i][j] = (v8f){0.f, 0.f, 0.f, 0.f, 0.f, 0.f, 0.f, 0.f};

    for (int k = 0; k < D; k += 4) {
        const size_t kp = (size_t)((k + koff) >> 1) * (2 * D);
        v2f bv1[NB], bv2[NB];
        #pragma unroll
        for (int j = 0; j < NB; ++j) {
            bv1[j] = *(const v2f*)(B1p + kp + 2 * ncol[j]);   // b64, 8B aligned
            bv2[j] = *(const v2f*)(B2p + kp + 2 * ncol[j]);
        }
        #pragma unroll
        for (int i = 0; i < MB; ++i) {
            v2f av1 = *(const v2f*)(a1[i] + k + koff);        // b64, 8B aligned
            v2f av2 = *(const v2f*)(a2[i] + k + koff);
            #pragma unroll
            for (int j = 0; j < NB; ++j) {
                acc[i][j] = __builtin_amdgcn_wmma_f32_16x16x4_f32(
                    false, av1, false, bv1[j], (short)0, acc[i][j], false, false);
                acc[i][j] = __builtin_amdgcn_wmma_f32_16x16x4_f32(
                    false, av2, false, bv2[j], (short)0, acc[i][j], false, false);
            }
        }
    }

    // Epilogue: C/D layout -> VGPR r = row 16*mt + r + 8*(lane>=16), col ncol[j]
    #pragma unroll
    for (int j = 0; j < NB; ++j) {
        const float bn = bias[ncol[j]];
        #pragma unroll
        for (int i = 0; i < MB; ++i) {
            if (mblk * MB + i < mtiles) {    // wave-uniform store guard
                const int rbase = mt[i] * 16 + ((lane >> 4) << 3);
                #pragma unroll
                for (int r = 0; r < 8; ++r) {
                    float v = acc[i][j][r] + bn;
                    v = v > 0.0f ? v : 0.0f;
                    Outp[(size_t)(rbase + r) * D + ncol[j]] = v;
                }
            }
        }
    }
}

// ---------------- row-wise l2 normalize ----------------
__global__ void rownorm_kernel(const float* __restrict__ In,
                               float* __restrict__ Outp) {
    int row = blockIdx.x;
    int t   = threadIdx.x;
    float v = In[row * D + t];
    __shared__ float red[D];
    red[t] = v * v;
    __syncthreads();
    for (int s = D / 2; s > 0; s >>= 1) {
        if (t < s) red[t] += red[t + s];
        __syncthreads();
    }
    float scale = 1.0f / fmaxf(sqrtf(red[0]), 1e-12f);
    Outp[row * D + t] = v * scale;
}

// ---------------- decode: Q[b] = relu(h[action[b]] * (hs[b].W4)) . W5 --------
__global__ void decode_kernel(const int* __restrict__ action_idx,
                              const float* __restrict__ h,
                              const float* __restrict__ hs,
                              const float* __restrict__ W4,
                              const float* __restrict__ W5,
                              float* __restrict__ Q) {
    int b = blockIdx.x;
    int t = threadIdx.x;
    __shared__ float red[D];

    red[t] = hs[b * D + t] * W4[t];
    __syncthreads();
    for (int s = D / 2; s > 0; s >>= 1) {
        if (t < s) red[t] += red[t + s];
        __syncthreads();
    }
    float sv = red[0];
    __syncthreads();

    int a   = action_idx[b];
    float z = h[a * D + t] * sv;
    z = z > 0.0f ? z : 0.0f;
    red[t] = z * W5[t];
    __syncthreads();
    for (int s = D / 2; s > 0; s >>= 1) {
        if (t < s) red[t] += red[t + s];
        __syncthreads();
    }
    if (t == 0) Q[b] = red[0];
}

extern "C" void kernel_launch(void* const* d_in, const int* in_sizes, int n_in,
                              void* d_out, int out_size, void* d_ws, size_t ws_size,
                              hipStream_t stream) {
    const int*   edge_src = (const int*)  d_in[0];
    const int*   edge_dst = (const int*)  d_in[1];
    const float* edge_w   = (const float*)d_in[2];
    const int*   batch    = (const int*)  d_in[3];
    const int*   act_idx  = (const int*)  d_in[4];
    const float* X        = (const float*)d_in[5];
    const float* Xs       = (const float*)d_in[6];
    const float* W1       = (const float*)d_in[7];
    const float* W2       = (const float*)d_in[8];
    const float* W3       = (const float*)d_in[9];
    const float* linW     = (const float*)d_in[10];
    const float* linb     = (const float*)d_in[11];
    const float* W4       = (const float*)d_in[12];
    const float* W5       = (const float*)d_in[13];
    float*       Q        = (float*)d_out;

    const int E = in_sizes[0];
    const int N = in_sizes[3];
    const int B = in_sizes[4];
    const int F = in_sizes[7] / D;

    // Workspace carve-up (floats)
    float* base  = (float*)d_ws;
    float* h     = base;                 // [N, D]
    float* htmp  = h    + (size_t)N * D; // [N, D] pre-norm
    float* h_nv  = htmp + (size_t)N * D; // [N, D]
    float* hs    = h_nv + (size_t)N * D; // [B, D]
    float* hstmp = hs    + (size_t)B * D;
    float* hs_nv = hstmp + (size_t)B * D;
    float* M2    = hs_nv + (size_t)B * D; // [D, D] packed
    float* M3    = M2 + D * D;            // [D, D] packed

    // Fold lin_W into W2/W3 and pack for WMMA B-operand b64 loads.
    premul_pack_kernel<<<D, D, 0, stream>>>(W2, linW,         M2);
    premul_pack_kernel<<<D, D, 0, stream>>>(W3, linW + D * D, M3);

    // Encode
    encode_kernel<<<N, D, 0, stream>>>(X,  W1, h,  F);
    encode_kernel<<<B, D, 0, stream>>>(Xs, W1, hs, F);

    const int mtN = N / 16;                       // 1250
    const int mtB = B / 16;                       // 4
    const int gemmBlocksN = (mtN + MB - 1) / MB;  // one block per 64-row m-block
    const int gemmBlocksB = (mtB + MB - 1) / MB;  // 1

    for (int depth = 0; depth < 3; ++depth) {
        zero_kernel<<<(N * D + 255) / 256, 256, 0, stream>>>(h_nv,  N * D);
        zero_kernel<<<(B * D + 255) / 256, 256, 0, stream>>>(hs_nv, B * D);

        edge_agg_kernel <<<E, D, 0, stream>>>(edge_src, edge_dst, edge_w, h, h_nv);
        batch_agg_kernel<<<N, D, 0, stream>>>(batch, h, hs_nv);

        gemm2_relu_kernel<<<gemmBlocksN, 256, 0, stream>>>(h,  h_nv,  M2, M3, linb, htmp,  mtN);
        gemm2_relu_kernel<<<gemmBlocksB, 256, 0, stream>>>(hs, hs_nv, M2, M3, linb, hstmp, mtB);

        rownorm_kernel<<<N, D, 0, stream>>>(htmp,  h);
        rownorm_kernel<<<B, D, 0, stream>>>(hstmp, hs);
    }

    decode_kernel<<<B, D, 0, stream>>>(act_idx, h, hs, W4, W5, Q);
}